// Backward_60902636258003
// MI455X (gfx1250) — compile-verified
//
#include <hip/hip_runtime.h>
#include <stdint.h>

// Bilinear warp (grid_sample, align_corners=True, border clamp) for
//   inp  [B=8, C=64, H=256, W=448] f32
//   flow [B, 2, H, W] f32  (x-displacement plane, then y-displacement plane)
//
// Memory-bound gather: ~0.48 GB minimum traffic -> ~20us floor @ 23.3 TB/s.
// Round-2 changes vs round-1:
//  * Hot-loop addressing rewritten as  uniform(SGPR) channel base
//    + zext(32-bit VGPR byte offset)  so the backend emits saddr-form
//    global_load_b64 / global_store_b128 with NO per-iteration 64-bit VALU
//    address math (channel advance is 1 scalar add).
//  * Output stores and flow loads are non-temporal (TH=NT): output is
//    write-once and flow read-once, so keep the 192MB L2 dedicated to the
//    input tensor, whose overlapping bilinear taps are the only reuse.

typedef float v2f __attribute__((ext_vector_type(2)));
typedef float v4f __attribute__((ext_vector_type(4)));
// 4-byte-aligned 2-float vector: keeps a b64 load even when the tap base is
// odd (DWORD-aligned only) -- verified in round 1 to lower to global_load_b64.
typedef v2f v2f_a4 __attribute__((aligned(4)));

static __device__ __forceinline__ v2f load_pair(const char* base, uint32_t boff) {
    return *reinterpret_cast<const v2f_a4*>(base + (size_t)boff);
}

__global__ __launch_bounds__(256) void warp_bilinear_kernel(
    const float* __restrict__ inp,
    const float* __restrict__ flow,
    float* __restrict__ out)
{
    constexpr int B = 8, C = 64, H = 256, W = 448;
    constexpr int HW = H * W;                 // 114,688
    constexpr int CHW = C * HW;               // 7,340,032

    const int t  = blockIdx.x * 256 + threadIdx.x;   // one thread = 4 pixels
    const int p4 = t * 4;                            // flat pixel index (b,y,x)
    const int b  = p4 / HW;
    const int r  = p4 - b * HW;
    const int y  = r / W;
    const int x  = r - y * W;                        // x % 4 == 0

    // ---- flow loads: two b128 NT loads (16B aligned, read-once) ----
    const float* fp = flow + (size_t)b * 2 * HW + (size_t)y * W + x;
    const v4f fx = __builtin_nontemporal_load(reinterpret_cast<const v4f*>(fp));
    const v4f fy = __builtin_nontemporal_load(reinterpret_cast<const v4f*>(fp + HW));

    // ---- per-pixel setup (done once, reused for all 64 channels) ----
    const float inv_hw = 2.0f / (float)(W - 1);      // flow normalization
    const float inv_hh = 2.0f / (float)(H - 1);
    const float gy = -1.0f + (float)y * inv_hh;      // linspace(-1,1,H)[y]

    // Per-thread part of every address fits in a 32-bit byte offset:
    // (b*CHW + off) * 4  <=  (7*7,340,032 + 114,688)*4 ~= 206 MB < 2^31.
    const uint32_t base_el = (uint32_t)b * (uint32_t)CHW;

    float    w00[4], w01[4], w10[4], w11[4];
    uint32_t boff0[4], boff1[4];

    #pragma unroll
    for (int i = 0; i < 4; ++i) {
        const float gx = -1.0f + (float)(x + i) * inv_hw;
        // normalize, add, clamp to [-1,1]  (matches reference exactly)
        const float sx = fminf(fmaxf(gx + fx[i] * inv_hw, -1.0f), 1.0f);
        const float sy = fminf(fmaxf(gy + fy[i] * inv_hh, -1.0f), 1.0f);
        // to pixel coords (align_corners=True): in [0, W-1] / [0, H-1]
        const float ix = (sx + 1.0f) * (float)(W - 1) * 0.5f;
        const float iy = (sy + 1.0f) * (float)(H - 1) * 0.5f;

        const float x0f = floorf(ix);
        const float y0f = floorf(iy);
        float wx1 = ix - x0f;
        float wy1 = iy - y0f;
        float wx0 = 1.0f - wx1;
        const float wy0 = 1.0f - wy1;

        int x0i = min(max((int)x0f, 0), W - 1);
        int y0i = min(max((int)y0f, 0), H - 1);
        const int y1i = min(y0i + 1, H - 1);

        // Horizontal taps are adjacent (x1 = x0+1) except at the clamped
        // border ix == W-1 exactly, where wx1 == 0. Shift the pair base to
        // W-2 and remap weights so pair.y carries the full weight:
        //   value = 0*row[W-2] + wx0*row[W-1]   (wx0 == 1 here).
        // Exact, branch-free in the hot loop, and removes the 4B OOB read at
        // the tensor's last element.
        int xb = x0i;
        if (x0i == W - 1) { xb = W - 2; wx1 = wx0; wx0 = 0.0f; }

        boff0[i] = (base_el + (uint32_t)(y0i * W + xb)) * 4u;
        boff1[i] = (base_el + (uint32_t)(y1i * W + xb)) * 4u;
        w00[i] = wy0 * wx0;  w01[i] = wy0 * wx1;
        w10[i] = wy1 * wx0;  w11[i] = wy1 * wx1;
    }

    const uint32_t obyte = (base_el + (uint32_t)(y * W + x)) * 4u;

    // ---- channel loop: 8 saddr gather loads + 1 b128 NT store / channel ----
    // Channel base is wave-uniform -> SGPR; all divergence lives in the
    // 32-bit byte offsets computed above.
    const char* pci = reinterpret_cast<const char*>(inp);
    char*       oci = reinterpret_cast<char*>(out);

    #pragma unroll 2
    for (int c = 0; c < C; ++c) {
        v4f res;
        #pragma unroll
        for (int i = 0; i < 4; ++i) {
            const v2f p0 = load_pair(pci, boff0[i]);  // (v00, v01)
            const v2f p1 = load_pair(pci, boff1[i]);  // (v10, v11)
            res[i] = w00[i] * p0.x + w01[i] * p0.y
                   + w10[i] * p1.x + w11[i] * p1.y;
        }
        __builtin_nontemporal_store(
            res, reinterpret_cast<v4f*>(oci + (size_t)obyte));  // 16B aligned
        pci += (size_t)HW * 4;   // uniform: 1 scalar add per channel
        oci += (size_t)HW * 4;
    }
}

extern "C" void kernel_launch(void* const* d_in, const int* in_sizes, int n_in,
                              void* d_out, int out_size, void* d_ws, size_t ws_size,
                              hipStream_t stream) {
    (void)in_sizes; (void)n_in; (void)out_size; (void)d_ws; (void)ws_size;
    const float* inp  = (const float*)d_in[0];   // [8,64,256,448]
    const float* flow = (const float*)d_in[1];   // [8,2,256,448]
    float*       out  = (float*)d_out;           // [8,64,256,448]

    constexpr int B = 8, H = 256, W = 448;
    constexpr int total_pix = B * H * W;         // 917,504
    constexpr int threads   = total_pix / 4;     // 229,376 (4 pixels/thread)
    constexpr int block     = 256;               // 8 wave32s per block
    constexpr int grid      = threads / block;   // 896 blocks, no tail
    static_assert(total_pix % 4 == 0 && threads % block == 0, "exact tiling");

    warp_bilinear_kernel<<<grid, block, 0, stream>>>(inp, flow, out);
}